// SpatialSmoothing_28595892257280
// MI455X (gfx1250) — compile-verified
//
#include <hip/hip_runtime.h>
#include <cstdint>

// 3x3 median filter (reflect pad) + clamp[0,1] over (32,3,512,512) f32.
// Memory-bound: ~200MB traffic -> ~8.6us floor @ 23.3 TB/s.
// CDNA5 path: async global->LDS DMA (ASYNCcnt) for the tile stage,
// 19-exchange median-of-9 network on wave32 VALU, NT 128-bit stores.

namespace {

constexpr int Himg = 512;
constexpr int Wimg = 512;
constexpr int TW = 64;        // tile width  (output pixels)
constexpr int TH = 16;        // tile height (output pixels)
constexpr int PW = TW + 2;    // padded tile width  = 66
constexpr int PH = TH + 2;    // padded tile height = 18
constexpr int LS = 67;        // LDS row stride (odd -> bank-conflict-free)
constexpr int NLOAD = PW * PH;    // 1188 elements to stage
constexpr int NTHREADS = 256;     // 8 wave32
constexpr int NPASS = (NLOAD + NTHREADS - 1) / NTHREADS;  // 5

typedef float v4f __attribute__((ext_vector_type(4)));

__device__ __forceinline__ void sort2(float& a, float& b) {
    float t = fminf(a, b);
    b = fmaxf(a, b);
    a = t;
}

__global__ __launch_bounds__(NTHREADS)
void median3x3_async(const float* __restrict__ x, float* __restrict__ y) {
    __shared__ float tile[PH * LS];

    const int t = threadIdx.x;
    const int x0 = blockIdx.x * TW;
    const int y0 = blockIdx.y * TH;
    const size_t planeOff = (size_t)blockIdx.z * (size_t)(Himg * Wimg);
    const float* __restrict__ src = x + planeOff;
    float* __restrict__ dst = y + planeOff;

    // ---- Stage 1: async DMA reflect-padded (18 x 66) tile into LDS ------
    // Uniform 5 passes (index clamped: duplicate lanes rewrite the same LDS
    // word with the same value -> benign, no divergent tail, no EXEC churn).
#pragma unroll
    for (int k = 0; k < NPASS; ++k) {
        int i = t + k * NTHREADS;
        i = (i < NLOAD) ? i : (NLOAD - 1);
        const int r = i / PW;
        const int c = i - r * PW;
        int gr = y0 + r - 1;
        int gc = x0 + c - 1;
        gr = (gr < 0) ? -gr : ((gr >= Himg) ? (2 * Himg - 2 - gr) : gr);
        gc = (gc < 0) ? -gc : ((gc >= Wimg) ? (2 * Wimg - 2 - gc) : gc);
        const unsigned long long gaddr =
            (unsigned long long)(uintptr_t)(src + (gr * Wimg + gc));
        const uint32_t lds = (uint32_t)(uintptr_t)&tile[r * LS + c];
        asm volatile("global_load_async_to_lds_b32 %0, %1, off"
                     :: "v"(lds), "v"(gaddr)
                     : "memory");
    }
    // Each wave drains its own async queue; the workgroup barrier then makes
    // all waves' DMA'd LDS contents visible to every reader.
    asm volatile("s_wait_asynccnt 0" ::: "memory");
    __syncthreads();

    // ---- Stage 2: 4 output pixels per thread from a shared 3x6 window ---
    const int row = t >> 4;          // 0..15
    const int col = (t & 15) * 4;    // 0..60, 16B-aligned output groups

    float w[3][6];
#pragma unroll
    for (int rr = 0; rr < 3; ++rr)
#pragma unroll
        for (int cc = 0; cc < 6; ++cc)
            w[rr][cc] = tile[(row + rr) * LS + (col + cc)];

    v4f res;
#pragma unroll
    for (int p = 0; p < 4; ++p) {
        float p0 = w[0][p], p1 = w[0][p + 1], p2 = w[0][p + 2];
        float p3 = w[1][p], p4 = w[1][p + 1], p5 = w[1][p + 2];
        float p6 = w[2][p], p7 = w[2][p + 1], p8 = w[2][p + 2];
        // Classic 19-exchange median-of-9 network; median lands in p4.
        sort2(p1, p2); sort2(p4, p5); sort2(p7, p8);
        sort2(p0, p1); sort2(p3, p4); sort2(p6, p7);
        sort2(p1, p2); sort2(p4, p5); sort2(p7, p8);
        sort2(p0, p3); sort2(p5, p8); sort2(p4, p7);
        sort2(p3, p6); sort2(p1, p4); sort2(p2, p5);
        sort2(p4, p7); sort2(p4, p2); sort2(p6, p4);
        sort2(p4, p2);
        res[p] = fminf(fmaxf(p4, 0.0f), 1.0f);   // clip to [0,1]
    }

    // Output is write-once/never re-read: non-temporal 128-bit store keeps
    // the 192MB L2 populated with reusable input (halo rows) instead.
    v4f* op = reinterpret_cast<v4f*>(dst + (size_t)(y0 + row) * Wimg + x0 + col);
    __builtin_nontemporal_store(res, op);
}

} // namespace

extern "C" void kernel_launch(void* const* d_in, const int* in_sizes, int n_in,
                              void* d_out, int out_size, void* d_ws, size_t ws_size,
                              hipStream_t stream) {
    const float* x = (const float*)d_in[0];
    float* y = (float*)d_out;
    const int planes = in_sizes[0] / (Himg * Wimg);   // B*C = 96
    dim3 grid(Wimg / TW, Himg / TH, planes);          // 8 x 32 x 96
    dim3 block(NTHREADS);
    hipLaunchKernelGGL(median3x3_async, grid, block, 0, stream, x, y);
}